// RegHD_AR_90812788507174
// MI455X (gfx1250) — compile-verified
//
#include <hip/hip_runtime.h>

typedef __attribute__((ext_vector_type(2))) float v2f;
typedef __attribute__((ext_vector_type(4))) float v4f;
typedef __attribute__((ext_vector_type(8))) float v8f;

#define S  1024
#define DN 65536
#define TD 64
#define TS 64
#define NCHUNK (S / TS)

__global__ __launch_bounds__(256) void reghd_main(
    const float* __restrict__ x,
    const float* __restrict__ weight,
    const float* __restrict__ bias,
    const float* __restrict__ alpha,
    float* __restrict__ q_out,    // d_out + 1
    float* __restrict__ hvs_out)  // d_out + 1 + DN, layout hvs[s*DN + d]
{
    __shared__ float E[TD][TS + 1];   // enc tile; stride 65 == 1 (mod 64 banks)
    __shared__ float x_lds[S];
    __shared__ float a_lds[S];
    __shared__ float red[2 * TD];     // per-K-half bundled partials

    const int tid = threadIdx.x;
    const int d0  = blockIdx.x * TD;

    // Preload x and alpha (4 KB each) into LDS once.
    #pragma unroll
    for (int j = 0; j < S / 256; ++j) {
        int idx = tid + 256 * j;
        x_lds[idx] = x[idx];
        a_lds[idx] = alpha[idx];
    }

    // WMMA role decomposition: 8 waves = 4 d-tiles (16 rows) x 2 K-halves (32 each).
    const int lane   = tid & 31;
    const int wave   = tid >> 5;
    const int d_tile = wave & 3;
    const int k_half = wave >> 2;
    const int mrow   = lane & 15;
    const int hi     = lane >> 4;     // 0: K pair {k,k+1}; 1: K pair {k+2,k+3}

    v8f acc = {0.f, 0.f, 0.f, 0.f, 0.f, 0.f, 0.f, 0.f};

    // Phase A mapping: 4 consecutive s per thread (b128 global loads)
    const int sq     = (tid & 15) * 4;   // s offset within chunk, 16B aligned
    const int d_sub  = tid >> 4;         // row base (0..15), rows d_sub + 16*i

    // Write-out mapping: 4 consecutive d per thread (b128 global stores)
    const int dq     = (tid & 15) * 4;   // d offset within tile, 16B aligned
    const int s_subW = tid >> 4;         // s base (0..15), s = s_subW + 16*i

    for (int c = 0; c < NCHUNK; ++c) {
        const int s0 = c * TS;
        __syncthreads();              // protect E reuse (also covers preload at c==0)

        // --- Phase A: elementwise encode, vectorized loads, stage in LDS ---
        #pragma unroll
        for (int i = 0; i < 4; ++i) {
            int dl = d_sub + 16 * i;
            size_t g = (size_t)(d0 + dl) * S + (s0 + sq);   // 16B aligned
            v4f w4 = *(const v4f*)&weight[g];
            v4f b4 = *(const v4f*)&bias[g];
            #pragma unroll
            for (int k = 0; k < 4; ++k) {
                float p = x_lds[s0 + sq + k] * w4[k];
                E[dl][sq + k] = __cosf(p + b4[k]) * __sinf(p);
            }
        }
        __syncthreads();

        // --- Phase B(a): WMMA mat-vec partial: acc[d] += enc[d,s] * alpha[s] ---
        // A: 16x4 f32 tile (lane m holds row m; VGPR0/1 = K pair, hi selects K+2).
        // B: 4x16 with every column = alpha segment -> all D columns identical.
        #pragma unroll
        for (int kk = 0; kk < 8; ++kk) {
            int k = k_half * 32 + kk * 4 + 2 * hi;
            v2f a, b;
            a.x = E[d_tile * 16 + mrow][k];
            a.y = E[d_tile * 16 + mrow][k + 1];
            b.x = a_lds[s0 + k];
            b.y = a_lds[s0 + k + 1];
            acc = __builtin_amdgcn_wmma_f32_16x16x4_f32(
                false, a, false, b, (short)0, acc, false, false);
        }

        // --- Phase B(b): transposed write-out, b128 stores ---
        // Lane gathers an LDS column of 4 (stride 65 -> distinct banks), stores 16B.
        #pragma unroll
        for (int i = 0; i < 4; ++i) {
            int sl = s_subW + 16 * i;
            v4f v;
            v.x = E[dq + 0][sl];
            v.y = E[dq + 1][sl];
            v.z = E[dq + 2][sl];
            v.w = E[dq + 3][sl];
            *(v4f*)&hvs_out[(size_t)(s0 + sl) * DN + (d0 + dq)] = v;  // 16B aligned
        }
    }

    // Extract bundled partials. D layout: VGPR r, lanes 0-15 -> M=r, lanes 16-31 -> M=r+8.
    // All 16 columns are identical (B columns equal), so lane mrow==0 of each half suffices.
    if (mrow == 0) {
        #pragma unroll
        for (int r = 0; r < 8; ++r)
            red[k_half * TD + d_tile * 16 + hi * 8 + r] = acc[r];
    }
    __syncthreads();

    if (tid < TD) {
        float bundled = red[tid] + red[TD + tid];
        q_out[d0 + tid] = bundled > 0.f ? 1.f : -1.f;
    }
}

// Deterministic single-block dot product: model_result = q . M
__global__ __launch_bounds__(1024) void reghd_dot(
    const float* __restrict__ q, const float* __restrict__ M,
    float* __restrict__ out)
{
    __shared__ float red[1024];
    int tid = threadIdx.x;
    float s = 0.f;
    for (int i = tid; i < DN; i += 1024) s += q[i] * M[i];
    red[tid] = s;
    __syncthreads();
    for (int ofs = 512; ofs > 0; ofs >>= 1) {
        if (tid < ofs) red[tid] += red[tid + ofs];
        __syncthreads();
    }
    if (tid == 0) out[0] = red[0];
}

extern "C" void kernel_launch(void* const* d_in, const int* in_sizes, int n_in,
                              void* d_out, int out_size, void* d_ws, size_t ws_size,
                              hipStream_t stream) {
    const float* x      = (const float*)d_in[0];
    const float* weight = (const float*)d_in[1];
    const float* bias   = (const float*)d_in[2];
    const float* alpha  = (const float*)d_in[3];
    const float* M      = (const float*)d_in[4];
    // d_in[5] = ts, unused (alpha row already selected by harness)

    float* out = (float*)d_out;
    float* q   = out + 1;
    float* hvs = out + 1 + DN;

    reghd_main<<<DN / TD, 256, 0, stream>>>(x, weight, bias, alpha, q, hvs);
    reghd_dot<<<1, 1024, 0, stream>>>(q, M, out);
}